// GAT2_48524540510783
// MI455X (gfx1250) — compile-verified
//
#include <hip/hip_runtime.h>
#include <math.h>

// Problem constants (match reference)
#define NN   20000
#define INC  256
#define OUTC 64
#define HHN  4
#define M1C  128
#define M2C  64
#define E1N  200000
#define E2N  400000

typedef float v2f __attribute__((ext_vector_type(2)));
typedef float v8f __attribute__((ext_vector_type(8)));

// ---------------------------------------------------------------------------
// FP32 WMMA GEMM: C[M,Ncol] = A[M,K] @ B[K,Ncol] (+ bias[col] if non-null)
// One wave -> one 16x64 strip (4 x 16x16 tiles), K stepped by 4 via
// V_WMMA_F32_16X16X4_F32. The A fragment is reused across the 4 column tiles
// (4 WMMA per 10 loaded floats per lane per K-step).
// Requires M%16==0, Ncol%64==0, K%4==0 (true for all GEMMs here).
// Fragment layouts per CDNA5 ISA 7.12.2:
//   A: lane L -> row m=L&15, ka = 2*(L>>4);  a.x=A[m][k0+ka], a.y=A[m][k0+ka+1]
//   B: lane L -> col n=L&15;                 b.x=B[k0+ka][n], b.y=B[k0+ka+1][n]
//   D: VGPR i -> C[row0 + 8*(L>>4) + i][n]
// ---------------------------------------------------------------------------
__global__ __launch_bounds__(256) void k_wmma_gemm_f32(
    const float* __restrict__ A, const float* __restrict__ B,
    const float* __restrict__ bias, float* __restrict__ C,
    int M, int K, int Ncol)
{
  const int wave = threadIdx.x >> 5;
  const int lane = threadIdx.x & 31;
  const int colGroups = Ncol >> 6;                 // 64-wide column groups
  const long long grp  = (long long)blockIdx.x * 8 + wave;
  const long long nGrp = (long long)(M >> 4) * colGroups;
  if (grp >= nGrp) return;  // wave-uniform exit: EXEC all-ones for WMMA

  const int row0 = (int)(grp / colGroups) << 4;
  const int col0 = (int)(grp % colGroups) << 6;
  const int mn = lane & 15;   // A row within tile / B+C col within tile
  const int kh = lane >> 4;   // K-half selector

  const float* __restrict__ Arow = A + (long long)(row0 + mn) * K;
  const float* __restrict__ Bp   = B + col0 + mn;

  v8f acc0 = {0.f,0.f,0.f,0.f,0.f,0.f,0.f,0.f};
  v8f acc1 = acc0, acc2 = acc0, acc3 = acc0;

  for (int k0 = 0; k0 < K; k0 += 4) {
    const int ka = k0 + (kh << 1);
    v2f av;
    av.x = Arow[ka];
    av.y = Arow[ka + 1];
    const float* __restrict__ br0 = Bp + (long long)ka * Ncol;
    const float* __restrict__ br1 = Bp + (long long)(ka + 1) * Ncol;
    v2f bv0, bv1, bv2, bv3;
    bv0.x = br0[0];  bv0.y = br1[0];
    bv1.x = br0[16]; bv1.y = br1[16];
    bv2.x = br0[32]; bv2.y = br1[32];
    bv3.x = br0[48]; bv3.y = br1[48];
    acc0 = __builtin_amdgcn_wmma_f32_16x16x4_f32(false, av, false, bv0, (short)0, acc0, false, false);
    acc1 = __builtin_amdgcn_wmma_f32_16x16x4_f32(false, av, false, bv1, (short)0, acc1, false, false);
    acc2 = __builtin_amdgcn_wmma_f32_16x16x4_f32(false, av, false, bv2, (short)0, acc2, false, false);
    acc3 = __builtin_amdgcn_wmma_f32_16x16x4_f32(false, av, false, bv3, (short)0, acc3, false, false);
  }

  const int rbase = row0 + (kh << 3);
  float* __restrict__ Cp = C + (long long)rbase * Ncol + col0 + mn;
  if (bias) {
    const float b0 = bias[col0 + mn];
    const float b1 = bias[col0 + mn + 16];
    const float b2 = bias[col0 + mn + 32];
    const float b3 = bias[col0 + mn + 48];
    #pragma unroll
    for (int i = 0; i < 8; ++i) {
      float* r = Cp + (long long)i * Ncol;
      r[0]  = acc0[i] + b0;
      r[16] = acc1[i] + b1;
      r[32] = acc2[i] + b2;
      r[48] = acc3[i] + b3;
    }
  } else {
    #pragma unroll
    for (int i = 0; i < 8; ++i) {
      float* r = Cp + (long long)i * Ncol;
      r[0]  = acc0[i];
      r[16] = acc1[i];
      r[32] = acc2[i];
      r[48] = acc3[i];
    }
  }
}

// ---------------------------------------------------------------------------
// Small utility / pointwise kernels
// ---------------------------------------------------------------------------
__global__ void k_fill_f32(float* p, float v, long long n) {
  long long i = (long long)blockIdx.x * blockDim.x + threadIdx.x;
  if (i < n) p[i] = v;
}
__global__ void k_fill_u32(unsigned* p, unsigned v, long long n) {
  long long i = (long long)blockIdx.x * blockDim.x + threadIdx.x;
  if (i < n) p[i] = v;
}

// Monotone float<->uint mapping so atomicMax(u32) implements float max.
__device__ __forceinline__ unsigned f32key(float f) {
  unsigned u = __float_as_uint(f);
  return (u & 0x80000000u) ? ~u : (u | 0x80000000u);
}
__device__ __forceinline__ float keyf32(unsigned k) {
  unsigned u = (k & 0x80000000u) ? (k & 0x7fffffffu) : ~k;
  return __uint_as_float(u);
}
#define NEG_INF_KEY 0x007FFFFFu  // f32key(-inf)

__device__ __forceinline__ float eluf(float v) {
  return v > 0.f ? v : (expf(v) - 1.0f);
}

// ssrc[n,h] = sum_c h[n,h,c]*a_s[h,c] ; sdst likewise (h layout: [N, H*Cper])
__global__ void k_node_logits(const float* __restrict__ h,
                              const float* __restrict__ a_s,
                              const float* __restrict__ a_d,
                              float* __restrict__ ssrc, float* __restrict__ sdst,
                              int n, int Cper)
{
  long long idx = (long long)blockIdx.x * blockDim.x + threadIdx.x;
  if (idx >= (long long)n * HHN) return;
  const int hd = (int)(idx % HHN);
  const long long node = idx / HHN;
  const float* __restrict__ hp = h + node * (long long)(HHN * Cper) + hd * Cper;
  const float* __restrict__ a1 = a_s + hd * Cper;
  const float* __restrict__ a2 = a_d + hd * Cper;
  float v1 = 0.f, v2 = 0.f;
  for (int c = 0; c < Cper; ++c) {
    const float hv = hp[c];
    v1 = fmaf(hv, a1[c], v1);
    v2 = fmaf(hv, a2[c], v2);
  }
  ssrc[idx] = v1;
  sdst[idx] = v2;
}

// e = leaky_relu(ssrc[src]+sdst[dst], 0.2); store e; segment max via atomicMax
__global__ void k_edge_logit_max(const int* __restrict__ src, const int* __restrict__ dst,
                                 const float* __restrict__ ssrc, const float* __restrict__ sdst,
                                 float* __restrict__ eb, unsigned* __restrict__ mk,
                                 long long EH)
{
  long long idx = (long long)blockIdx.x * blockDim.x + threadIdx.x;
  if (idx >= EH) return;
  const int hd = (int)(idx % HHN);
  const long long e = idx / HHN;
  const int s = src[e], d = dst[e];
  float v = ssrc[(long long)s * HHN + hd] + sdst[(long long)d * HHN + hd];
  v = v > 0.f ? v : 0.2f * v;
  eb[idx] = v;
  atomicMax(&mk[(long long)d * HHN + hd], f32key(v));
}

// ex = exp(e - m[dst]); e <- ex; den[dst] += ex
__global__ void k_edge_exp_sum(const int* __restrict__ dst, float* __restrict__ eb,
                               const unsigned* __restrict__ mk, float* __restrict__ den,
                               long long EH)
{
  long long idx = (long long)blockIdx.x * blockDim.x + threadIdx.x;
  if (idx >= EH) return;
  const int hd = (int)(idx % HHN);
  const long long e = idx / HHN;
  const int d = dst[e];
  const float m = keyf32(mk[(long long)d * HHN + hd]);
  const float ex = expf(eb[idx] - m);
  eb[idx] = ex;
  atomicAdd(&den[(long long)d * HHN + hd], ex);
}

// out[dst, col..col+3] += alpha * h[src, col..col+3], float4-vectorized gather.
// Cper % 4 == 0 so a float4 never crosses a head boundary.
__global__ void k_edge_scatter4(const int* __restrict__ src, const int* __restrict__ dst,
                                const float* __restrict__ eb, const float* __restrict__ den,
                                const float* __restrict__ h, float* __restrict__ out,
                                long long total4, int Ctot, int Cper)
{
  long long idx = (long long)blockIdx.x * blockDim.x + threadIdx.x;
  if (idx >= total4) return;
  const int c4groups = Ctot >> 2;
  const int col = (int)(idx % c4groups) << 2;
  const long long e = idx / c4groups;
  const int hd = col / Cper;
  const int s = src[e], d = dst[e];
  const float alpha = eb[e * HHN + hd] / (den[(long long)d * HHN + hd] + 1e-16f);
  const float4 hv = *(const float4*)(h + (long long)s * Ctot + col);
  float* __restrict__ op = out + (long long)d * Ctot + col;
  atomicAdd(op + 0, alpha * hv.x);
  atomicAdd(op + 1, alpha * hv.y);
  atomicAdd(op + 2, alpha * hv.z);
  atomicAdd(op + 3, alpha * hv.w);
}

// o <- elu(o + b[col])
__global__ void k_elu_bias(float* __restrict__ o, const float* __restrict__ b,
                           long long total, int Ctot)
{
  long long idx = (long long)blockIdx.x * blockDim.x + threadIdx.x;
  if (idx >= total) return;
  o[idx] = eluf(o[idx] + b[idx % Ctot]);
}

// out[n,c] = elu(mean_h(o[n,h,c]) + b[c])   (concat=False head-mean)
__global__ void k_mean_heads_elu(const float* __restrict__ o, const float* __restrict__ b,
                                 float* __restrict__ outp, int n, int Cper)
{
  long long idx = (long long)blockIdx.x * blockDim.x + threadIdx.x;
  if (idx >= (long long)n * Cper) return;
  const int c = (int)(idx % Cper);
  const long long node = idx / Cper;
  const float* __restrict__ op = o + node * (long long)(HHN * Cper) + c;
  float s = 0.f;
  #pragma unroll
  for (int h = 0; h < HHN; ++h) s += op[h * Cper];
  s = s * (1.0f / HHN) + b[c];
  outp[idx] = eluf(s);
}

// w1[n] = softmax over {dot(x1[n],g[0]), dot(x2[n],g[1])} -> first weight
__global__ void k_aggr_w(const float* __restrict__ x1, const float* __restrict__ x2,
                         const float* __restrict__ g, float* __restrict__ w,
                         int n, int C)
{
  long long node = (long long)blockIdx.x * blockDim.x + threadIdx.x;
  if (node >= n) return;
  const float* __restrict__ p1 = x1 + node * (long long)C;
  const float* __restrict__ p2 = x2 + node * (long long)C;
  float s1 = 0.f, s2 = 0.f;
  for (int c = 0; c < C; ++c) {
    s1 = fmaf(p1[c], g[c], s1);
    s2 = fmaf(p2[c], g[C + c], s2);
  }
  w[node] = 1.0f / (1.0f + expf(s2 - s1));
}

// out += w1*x1 + (1-w1)*x2   (out pre-filled with the linear branch)
__global__ void k_aggr_combine(const float* __restrict__ x1, const float* __restrict__ x2,
                               const float* __restrict__ w, float* __restrict__ out,
                               long long total, int C)
{
  long long idx = (long long)blockIdx.x * blockDim.x + threadIdx.x;
  if (idx >= total) return;
  const float wv = w[idx / C];
  out[idx] += wv * x1[idx] + (1.0f - wv) * x2[idx];
}

// ---------------------------------------------------------------------------
// Host orchestration
// ---------------------------------------------------------------------------
static inline unsigned cdiv(long long n, int b) { return (unsigned)((n + b - 1) / b); }

extern "C" void kernel_launch(void* const* d_in, const int* in_sizes, int n_in,
                              void* d_out, int out_size, void* d_ws, size_t ws_size,
                              hipStream_t stream)
{
  const float* x    = (const float*)d_in[0];
  const int*   adj1 = (const int*)d_in[1];
  const int*   adj2 = (const int*)d_in[2];
  const int *s1 = adj1, *dd1 = adj1 + E1N;
  const int *s2 = adj2, *dd2 = adj2 + E2N;

  struct GatP { const float *W, *as, *ad, *b; };
  int p = 3;
  auto getGat = [&]() {
    GatP g;
    g.W  = (const float*)d_in[p++]; g.as = (const float*)d_in[p++];
    g.ad = (const float*)d_in[p++]; g.b  = (const float*)d_in[p++];
    return g;
  };
  GatP c11 = getGat(), c12 = getGat(), c21 = getGat(), c22 = getGat(),
       c31 = getGat(), c32 = getGat();
  const float* g1 = (const float*)d_in[p++];
  const float* g2 = (const float*)d_in[p++];
  const float* g3 = (const float*)d_in[p++];
  const float* lin1W = (const float*)d_in[p++]; const float* lin1b = (const float*)d_in[p++];
  const float* lin2W = (const float*)d_in[p++]; const float* lin2b = (const float*)d_in[p++];
  const float* lin3W = (const float*)d_in[p++]; const float* lin3b = (const float*)d_in[p++];

  // Workspace carve-up (floats). Peak ~202 MB; hot set is largely L2-resident.
  float* ws = (float*)d_ws;
  size_t off = 0;
  float*    HB   = ws;        off += (size_t)NN * 512;   // x@W per branch
  float*    O1   = ws + off;  off += (size_t)NN * 512;   // branch-1 GAT output
  float*    O2   = ws + off;  off += (size_t)NN * 512;   // branch-2 GAT output
  float*    MIDA = ws + off;  off += (size_t)NN * 512;   // layer-1 result
  float*    MIDB = ws + off;  off += (size_t)NN * 256;   // layer-2 result
  float*    EB   = ws + off;  off += (size_t)E2N * HHN;  // per-edge e / ex
  float*    SS   = ws + off;  off += (size_t)NN * HHN;   // src logits
  float*    SD   = ws + off;  off += (size_t)NN * HHN;   // dst logits
  unsigned* MK   = (unsigned*)(ws + off); off += (size_t)NN * HHN; // seg-max keys
  float*    DEN  = ws + off;  off += (size_t)NN * HHN;   // softmax denom
  float*    WA   = ws + off;  off += (size_t)NN;         // aggr weight
  float*    NS1  = ws + off;  off += (size_t)NN * OUTC;  // layer-3 branch means
  float*    NS2  = ws + off;  off += (size_t)NN * OUTC;

  auto gemm = [&](const float* A, const float* B, const float* bias, float* C,
                  int M, int K, int Ncol) {
    long long grps = (long long)(M >> 4) * (Ncol >> 6);
    k_wmma_gemm_f32<<<dim3((unsigned)((grps + 7) / 8)), dim3(256), 0, stream>>>(
        A, B, bias, C, M, K, Ncol);
  };

  auto gat = [&](const float* xin, int Cin, int Cper, const GatP& q,
                 const int* src, const int* dst, int E, float* obuf) {
    const int Ctot = HHN * Cper;
    gemm(xin, q.W, nullptr, HB, NN, Cin, Ctot);
    const long long nh = (long long)NN * HHN;
    k_node_logits<<<cdiv(nh, 256), 256, 0, stream>>>(HB, q.as, q.ad, SS, SD, NN, Cper);
    k_fill_u32<<<cdiv(nh, 256), 256, 0, stream>>>(MK, NEG_INF_KEY, nh);
    k_fill_f32<<<cdiv(nh, 256), 256, 0, stream>>>(DEN, 0.f, nh);
    const long long no = (long long)NN * Ctot;
    k_fill_f32<<<cdiv(no, 256), 256, 0, stream>>>(obuf, 0.f, no);
    const long long eh = (long long)E * HHN;
    k_edge_logit_max<<<cdiv(eh, 256), 256, 0, stream>>>(src, dst, SS, SD, EB, MK, eh);
    k_edge_exp_sum<<<cdiv(eh, 256), 256, 0, stream>>>(dst, EB, MK, DEN, eh);
    const long long total4 = (long long)E * (Ctot >> 2);
    k_edge_scatter4<<<cdiv(total4, 256), 256, 0, stream>>>(src, dst, EB, DEN, HB, obuf,
                                                           total4, Ctot, Cper);
  };

  // ---- Layer 1: IN=256 -> H*M1=512 (concat) ----
  gat(x, INC, M1C, c11, s1, dd1, E1N, O1);
  k_elu_bias<<<cdiv((long long)NN * 512, 256), 256, 0, stream>>>(O1, c11.b, (long long)NN * 512, 512);
  gat(x, INC, M1C, c12, s2, dd2, E2N, O2);
  k_elu_bias<<<cdiv((long long)NN * 512, 256), 256, 0, stream>>>(O2, c12.b, (long long)NN * 512, 512);
  gemm(x, lin1W, lin1b, MIDA, NN, INC, 512);
  k_aggr_w<<<cdiv(NN, 256), 256, 0, stream>>>(O1, O2, g1, WA, NN, 512);
  k_aggr_combine<<<cdiv((long long)NN * 512, 256), 256, 0, stream>>>(O1, O2, WA, MIDA,
                                                                     (long long)NN * 512, 512);

  // ---- Layer 2: 512 -> H*M2=256 (concat) ----
  gat(MIDA, 512, M2C, c21, s1, dd1, E1N, O1);
  k_elu_bias<<<cdiv((long long)NN * 256, 256), 256, 0, stream>>>(O1, c21.b, (long long)NN * 256, 256);
  gat(MIDA, 512, M2C, c22, s2, dd2, E2N, O2);
  k_elu_bias<<<cdiv((long long)NN * 256, 256), 256, 0, stream>>>(O2, c22.b, (long long)NN * 256, 256);
  gemm(MIDA, lin2W, lin2b, MIDB, NN, 512, 256);
  k_aggr_w<<<cdiv(NN, 256), 256, 0, stream>>>(O1, O2, g2, WA, NN, 256);
  k_aggr_combine<<<cdiv((long long)NN * 256, 256), 256, 0, stream>>>(O1, O2, WA, MIDB,
                                                                     (long long)NN * 256, 256);

  // ---- Layer 3: 256 -> OUT=64 (head-mean) ----
  float* out = (float*)d_out;
  gat(MIDB, 256, OUTC, c31, s1, dd1, E1N, O1);
  k_mean_heads_elu<<<cdiv((long long)NN * OUTC, 256), 256, 0, stream>>>(O1, c31.b, NS1, NN, OUTC);
  gat(MIDB, 256, OUTC, c32, s2, dd2, E2N, O2);
  k_mean_heads_elu<<<cdiv((long long)NN * OUTC, 256), 256, 0, stream>>>(O2, c32.b, NS2, NN, OUTC);
  gemm(MIDB, lin3W, lin3b, out, NN, 256, OUTC);
  k_aggr_w<<<cdiv(NN, 256), 256, 0, stream>>>(NS1, NS2, g3, WA, NN, OUTC);
  k_aggr_combine<<<cdiv((long long)NN * OUTC, 256), 256, 0, stream>>>(NS1, NS2, WA, out,
                                                                      (long long)NN * OUTC, OUTC);
}